// CBFocalClassifierV0_34471407518073
// MI455X (gfx1250) — compile-verified
//
#include <hip/hip_runtime.h>
#include <cstdint>

#define TPB_ROW 1024
#define MAX_C   32000
#define TPB_CLS 256
#define MAX_B   4096

__device__ __forceinline__ float wave_max(float v) {
#pragma unroll
  for (int o = 16; o; o >>= 1) v = fmaxf(v, __shfl_xor(v, o, 32));
  return v;
}
__device__ __forceinline__ float wave_sum(float v) {
#pragma unroll
  for (int o = 16; o; o >>= 1) v += __shfl_xor(v, o, 32);
  return v;
}

// ---------------------------------------------------------------------------
// Kernel 1: one workgroup per row. Stage the 128KB row in LDS via CDNA5 async
// global->LDS copies (ASYNCcnt), then 3 LDS passes: max, sum(exp), focal sum.
// ---------------------------------------------------------------------------
__global__ __launch_bounds__(TPB_ROW) void row_focal_kernel(
    const float* __restrict__ pred, const int* __restrict__ target,
    float* __restrict__ ent, int Cn) {
  __shared__ __align__(16) float lds[MAX_C];     // 128000 B row stage
  __shared__ float red[TPB_ROW / 32];            // cross-wave reduction
  __shared__ float bc0;                          // broadcast slot

  const int tid = threadIdx.x;
  const int row = blockIdx.x;
  const float* rowp = pred + (size_t)row * (size_t)Cn;
  const int nvec = Cn >> 2;                      // float4 elements (8000)

  // ---- async global -> LDS staging (single HBM read of pred) ----
  uint32_t lds_base = (uint32_t)(uintptr_t)(&lds[0]);
  for (int i = tid; i < nvec; i += TPB_ROW) {
    uint32_t loff = lds_base + ((uint32_t)i << 4);  // LDS byte address
    uint32_t goff = (uint32_t)i << 4;               // byte offset from saddr
    asm volatile("global_load_async_to_lds_b128 %0, %1, %2"
                 :: "v"(loff), "v"(goff), "s"(rowp) : "memory");
  }
#if __has_builtin(__builtin_amdgcn_s_wait_asynccnt)
  __builtin_amdgcn_s_wait_asynccnt(0);
#else
  asm volatile("s_wait_asynccnt 0" ::: "memory");
#endif
  __syncthreads();

  const float4* lv = (const float4*)lds;
  const int w = tid >> 5, l = tid & 31;

  // ---- pass 1: row max ----
  float m = -INFINITY;
  for (int i = tid; i < nvec; i += TPB_ROW) {
    float4 v = lv[i];
    m = fmaxf(m, fmaxf(fmaxf(v.x, v.y), fmaxf(v.z, v.w)));
  }
  m = wave_max(m);
  if (l == 0) red[w] = m;
  __syncthreads();
  if (w == 0) {
    float x = red[l];
    x = wave_max(x);
    if (l == 0) bc0 = x;
  }
  __syncthreads();
  const float rmax = bc0;
  __syncthreads();

  // ---- pass 2: Z = sum exp(x - max) ----
  float s = 0.f;
  for (int i = tid; i < nvec; i += TPB_ROW) {
    float4 v = lv[i];
    s += __expf(v.x - rmax) + __expf(v.y - rmax) +
         __expf(v.z - rmax) + __expf(v.w - rmax);
  }
  s = wave_sum(s);
  if (l == 0) red[w] = s;
  __syncthreads();
  if (w == 0) {
    float x = red[l];
    x = wave_sum(x);
    if (l == 0) bc0 = x;
  }
  __syncthreads();
  const float Z = bc0;
  __syncthreads();
  const float logZ = __logf(Z);
  const float invZ = 1.0f / Z;

  // ---- pass 3: focal = sum (1-p)^2 * logp ----
  float f = 0.f;
  for (int i = tid; i < nvec; i += TPB_ROW) {
    float4 v = lv[i];
    float d, lp, p, om;
    d = v.x - rmax; lp = d - logZ; p = __expf(d) * invZ; om = 1.f - p; f += om * om * lp;
    d = v.y - rmax; lp = d - logZ; p = __expf(d) * invZ; om = 1.f - p; f += om * om * lp;
    d = v.z - rmax; lp = d - logZ; p = __expf(d) * invZ; om = 1.f - p; f += om * om * lp;
    d = v.w - rmax; lp = d - logZ; p = __expf(d) * invZ; om = 1.f - p; f += om * om * lp;
  }
  f = wave_sum(f);
  if (l == 0) red[w] = f;
  __syncthreads();
  if (w == 0) {
    float x = red[l];
    x = wave_sum(x);
    if (l == 0) ent[row] = (float)target[row] * x;  // ent_b = target_b * focal_b
  }
}

// ---------------------------------------------------------------------------
// Kernel 2: deterministic class aggregation (no atomics). Each thread owns one
// class, scans the whole batch from LDS, then block-reduce of w_c * cls_sum_c.
// ---------------------------------------------------------------------------
__global__ __launch_bounds__(TPB_CLS) void class_kernel(
    const float* __restrict__ ent, const int* __restrict__ target,
    float* __restrict__ partials, int Bn, int Cn) {
  __shared__ float sent[MAX_B];
  __shared__ int   stgt[MAX_B];
  __shared__ float red[TPB_CLS / 32];

  const int tid = threadIdx.x;
  for (int i = tid; i < Bn; i += TPB_CLS) {
    sent[i] = ent[i];
    stgt[i] = target[i];
  }
  __syncthreads();

  const int c = blockIdx.x * TPB_CLS + tid;
  float s = 0.f, cnt = 0.f;
  if (c < Cn) {
    for (int b = 0; b < Bn; ++b) {
      bool hit = (stgt[b] == c);
      s   += hit ? sent[b] : 0.f;
      cnt += hit ? 1.f : 0.f;
    }
  }
  const float beta = (float)(Bn - 1) / (float)Bn;
  const float wgt = (1.0f - beta) /
                    (1.0f - __expf(cnt * __logf(beta)) + 1e-6f);  // beta^cnt
  float val = (c < Cn) ? wgt * s : 0.f;

  val = wave_sum(val);
  const int w = tid >> 5, l = tid & 31;
  if (l == 0) red[w] = val;
  __syncthreads();
  if (w == 0) {
    float x = (l < TPB_CLS / 32) ? red[l] : 0.f;
    x = wave_sum(x);
    if (l == 0) partials[blockIdx.x] = x;
  }
}

// ---------------------------------------------------------------------------
// Kernel 3: final deterministic reduction of block partials, apply -1/B.
// ---------------------------------------------------------------------------
__global__ __launch_bounds__(256) void final_kernel(
    const float* __restrict__ partials, int n, float* __restrict__ out, int Bn) {
  __shared__ float red[8];
  const int tid = threadIdx.x;
  float v = 0.f;
  for (int i = tid; i < n; i += 256) v += partials[i];
  v = wave_sum(v);
  const int w = tid >> 5, l = tid & 31;
  if (l == 0) red[w] = v;
  __syncthreads();
  if (w == 0) {
    float x = (l < 8) ? red[l] : 0.f;
    x = wave_sum(x);
    if (l == 0) out[0] = (-1.0f / (float)Bn) * x;
  }
}

extern "C" void kernel_launch(void* const* d_in, const int* in_sizes, int n_in,
                              void* d_out, int out_size, void* d_ws, size_t ws_size,
                              hipStream_t stream) {
  const float* pred   = (const float*)d_in[0];
  const int*   target = (const int*)d_in[1];   // jax default x64-off: int32

  const int Bn = in_sizes[1];                  // 4096
  const int Cn = in_sizes[0] / Bn;             // 32000

  float* ent      = (float*)d_ws;              // Bn floats
  float* partials = ent + Bn;                  // nBlocks floats
  const int nClsBlocks = (Cn + TPB_CLS - 1) / TPB_CLS;

  row_focal_kernel<<<Bn, TPB_ROW, 0, stream>>>(pred, target, ent, Cn);
  class_kernel<<<nClsBlocks, TPB_CLS, 0, stream>>>(ent, target, partials, Bn, Cn);
  final_kernel<<<1, 256, 0, stream>>>(partials, nClsBlocks, (float*)d_out, Bn);
}